// SelfAttention_35871566856406
// MI455X (gfx1250) — compile-verified
//
#include <hip/hip_runtime.h>
#include <hip/hip_bf16.h>
#include <cstdint>
#include <cstddef>

// ---------------------------------------------------------------------------
// RoPE over fused QKV + transpose (b,s,h,d) -> (b,h,s,d), Q/K -> bf16, V fp32.
// Memory-bound: ~335 MB moved -> ~14 us floor at 23.3 TB/s.
// CDNA5 path used: GLOBAL_LOAD_ASYNC_TO_LDS_B128 (ASYNCcnt) to stage each
// 48 KB token into LDS, then conflict-free DS reads feed the rotary math.
// ---------------------------------------------------------------------------

#define BDIM 256
constexpr int B  = 2;
constexpr int S  = 2048;
constexpr int NH = 32;
constexpr int HD = 128;
constexpr int H3 = 3 * NH * HD;        // 12288 floats per token
constexpr int ITERS = (H3 / 4) / BDIM; // 12 float4s per thread

#if __has_builtin(__builtin_amdgcn_global_load_async_to_lds_b128)
#define USE_ASYNC_LDS 1
typedef int v4i __attribute__((ext_vector_type(4)));
typedef __attribute__((address_space(1))) v4i gv4i;
typedef __attribute__((address_space(3))) v4i lv4i;
#endif

__global__ __launch_bounds__(BDIM)
void rope_qkv_kernel(const float* __restrict__ qkv,
                     const int* __restrict__ pos,
                     __hip_bfloat16* __restrict__ qOut,
                     __hip_bfloat16* __restrict__ kOut,
                     float* __restrict__ vOut) {
  __shared__ __align__(16) float lds[H3];        // one full token: 48 KB
  __shared__ __align__(16) float tabC[2][32];    // cos tables: [pid, bid]
  __shared__ __align__(16) float tabS[2][32];    // sin tables

  const int tid = threadIdx.x;
  const int tok = blockIdx.x;           // 0 .. B*S-1
  const int b   = tok >> 11;            // tok / 2048
  const int s   = tok & 2047;

  const float* gsrc = qkv + (size_t)tok * H3;

  // ---- Stage token into LDS: fire-and-forget async b128 copies ----
#if defined(USE_ASYNC_LDS)
  #pragma unroll
  for (int it = 0; it < ITERS; ++it) {
    const int f = (it * BDIM + tid) * 4;  // float offset, 16B aligned
    __builtin_amdgcn_global_load_async_to_lds_b128(
        (gv4i*)(gsrc + f),
        (lv4i*)(lds + f),
        0, 0);
  }
#else
  for (int it = 0; it < ITERS; ++it) {
    const int f4 = it * BDIM + tid;
    ((float4*)lds)[f4] = ((const float4*)gsrc)[f4];
  }
#endif

  // ---- Overlap: build cos/sin tables (64 lanes, depends only on pos ids) ----
  if (tid < 64) {
    const int tbl = tid >> 5;           // 0 = pid table, 1 = bid table
    const int j   = tid & 31;           // frequency index 0..31
    const int p   = pos[((size_t)b * 2 + tbl) * S + s];
    // inv_freq[j] = 10000^(-j/32) = exp(-j * ln(10000)/32)
    const float inv = expf(-(float)j * (9.210340371976184f / 32.0f));
    const float ang = (float)p * inv;
    float sv, cv;
    sincosf(ang, &sv, &cv);
    tabC[tbl][j] = cv;
    tabS[tbl][j] = sv;
  }

#if defined(USE_ASYNC_LDS)
  #if __has_builtin(__builtin_amdgcn_s_wait_asynccnt)
  __builtin_amdgcn_s_wait_asynccnt(0);
  #else
  asm volatile("s_wait_asynccnt 0x0" ::: "memory");
  #endif
#endif
  __syncthreads();

  // ---- Rotary math from LDS, packed coalesced stores ----
  const int wave = tid >> 5;            // 0..7, each wave does 4 heads
  const int j    = tid & 31;            // lane: output dims {2j, 2j+1}
  const int d0   = 2 * j;
  const int half = j >> 4;              // which 32-half of the 64-dim block
  const int po   = half ? (d0 - 32) : (d0 + 32);  // rotate_half partner
  const float sgn = half ? 1.0f : -1.0f;
  const int i2   = j & 15;              // float2 index into tables

  auto ropeStore = [&](const float* xb, int tbl, __hip_bfloat16* outp) {
    const float2 x  = *(const float2*)(xb + d0);
    const float2 p  = *(const float2*)(xb + po);
    const float2 c  = ((const float2*)tabC[tbl])[i2];
    const float2 sn = ((const float2*)tabS[tbl])[i2];
    const float o0 = x.x * c.x + sgn * (p.x * sn.x);
    const float o1 = x.y * c.y + sgn * (p.y * sn.y);
    const unsigned lo = (unsigned)__bfloat16_as_ushort(__float2bfloat16(o0));
    const unsigned hi = (unsigned)__bfloat16_as_ushort(__float2bfloat16(o1));
    *(unsigned*)(outp + d0) = (hi << 16) | lo;   // dword store, 128B/wave
  };

  #pragma unroll
  for (int hh = 0; hh < 4; ++hh) {
    const int h = wave * 4 + hh;
    const float* base = lds + h * (3 * HD);  // [q(128) | k(128) | v(128)]
    const size_t row = (((size_t)b * NH + h) * S + s) * HD;

    // Q: dims 0..63 with pid table, dims 64..127 with bid table
    ropeStore(base,        0, qOut + row);
    ropeStore(base + 64,   1, qOut + row + 64);
    // K
    ropeStore(base + 128,  0, kOut + row);
    ropeStore(base + 192,  1, kOut + row + 64);
    // V: passthrough fp32, float2 per lane (256B/wave stores)
    *(float2*)(vOut + row + d0)      = *(const float2*)(base + 256 + d0);
    *(float2*)(vOut + row + 64 + d0) = *(const float2*)(base + 320 + d0);
  }
}

// ---------------------------------------------------------------------------
// d_out layout assumption for the tuple (q, k, v), flat in return order:
//   [ q: 16,777,216 bf16 ][ k: 16,777,216 bf16 ][ v: 16,777,216 fp32 ]
// ---------------------------------------------------------------------------
extern "C" void kernel_launch(void* const* d_in, const int* in_sizes, int n_in,
                              void* d_out, int out_size, void* d_ws, size_t ws_size,
                              hipStream_t stream) {
  (void)in_sizes; (void)n_in; (void)out_size; (void)d_ws; (void)ws_size;
  const float* qkv = (const float*)d_in[0];
  const int*   pos = (const int*)d_in[1];

  const size_t N = (size_t)B * NH * S * HD;   // 16,777,216 per tensor
  __hip_bfloat16* qOut = (__hip_bfloat16*)d_out;
  __hip_bfloat16* kOut = qOut + N;
  float*          vOut = (float*)(kOut + N);

  rope_qkv_kernel<<<dim3(B * S), dim3(BDIM), 0, stream>>>(qkv, pos, qOut, kOut, vOut);
}